// EntityMarker_44040594653559
// MI455X (gfx1250) — compile-verified
//
#include <hip/hip_runtime.h>

typedef __attribute__((ext_vector_type(2))) float v2f;
typedef __attribute__((ext_vector_type(8))) float v8f;

// One wave owns one 16-column strip of one (batch, head/tail) span.
// Span-sum over rows: D = A(ones,16x4) * B(rows,4x16) + C on the f32 WMMA
// pipe. B layout (32-bit, 4x16): lane = {khalf=lane>>4, n=lane&15},
// VGPR v holds K = khalf*2 + v, N = n. All D rows equal the column sums.
template <int H_C>
__global__ __launch_bounds__(256) void span_mean_wmma_kernel(
    const float* __restrict__ seq,   // [B, L, H] f32
    const int*   __restrict__ epos,  // [B, 4] i32
    float*       __restrict__ out,   // [2, B, H] f32 (head then tail)
    int B, int L, int Hdyn)
{
    const int H = (H_C > 0) ? H_C : Hdyn;

    const int lane  = threadIdx.x & 31;
    const int wave  = threadIdx.x >> 5;
    const int s     = blockIdx.y;        // 0 .. 2B-1
    const int b     = s >> 1;
    const int tail  = s & 1;

    const int n     = lane & 15;         // column within strip (N)
    const int khalf = lane >> 4;         // lanes 0-15 -> K{0,1}, 16-31 -> K{2,3}

    int col = blockIdx.x * 128 + wave * 16 + n;
    const bool valid = (col < H);
    if (!valid) col = H - 1;             // clamp address; result never stored

    // Span bounds, clamped exactly like the reference.
    const int* ep = epos + b * 4 + (tail ? 2 : 0);
    int p0 = ep[0], p1 = ep[1];
    int start = p0 < 0 ? 0 : (p0 > L - 1 ? L - 1 : p0);
    int e     = p1 > L - 1 ? L - 1 : p1;
    int end   = e < start ? start : e;
    const int count = end - start + 1;

    const size_t strideH = (size_t)H;
    const float* rowbase = seq + (size_t)b * (size_t)L * strideH + col; // row 0
    // Per-lane streaming pointer at row (start + khalf*2), column col.
    const float* p = rowbase + (size_t)(start + khalf * 2) * strideH;

    v2f a  = {1.0f, 1.0f};               // all-ones A matrix (16x4)
    v8f c0 = {}, c1 = {}, c2 = {}, c3 = {};

    // ---- Hot loop: 16 rows per iteration, no conditionals, immediate offsets.
    const int nfull = count >> 4;
#pragma unroll 1
    for (int i = 0; i < nfull; ++i) {
        v2f b0 = { p[0],             p[strideH] };
        v2f b1 = { p[4 * strideH],   p[5 * strideH] };
        v2f b2 = { p[8 * strideH],   p[9 * strideH] };
        v2f b3 = { p[12 * strideH],  p[13 * strideH] };
        p += 16 * strideH;
        c0 = __builtin_amdgcn_wmma_f32_16x16x4_f32(false, a, false, b0, (short)0, c0, false, false);
        c1 = __builtin_amdgcn_wmma_f32_16x16x4_f32(false, a, false, b1, (short)0, c1, false, false);
        c2 = __builtin_amdgcn_wmma_f32_16x16x4_f32(false, a, false, b2, (short)0, c2, false, false);
        c3 = __builtin_amdgcn_wmma_f32_16x16x4_f32(false, a, false, b3, (short)0, c3, false, false);
    }

    // ---- Masked tail (executed once): clamped addresses + selects, no branches.
    if (count & 15) {
        const int rbase = start + (nfull << 4) + khalf * 2;
        auto ldm = [&](int off) -> float {
            int r  = rbase + off;
            int rc = r > end ? end : r;                 // address always in-span
            float v = rowbase[(size_t)rc * strideH];
            return (r <= end) ? v : 0.0f;               // v_cndmask, no exec branch
        };
        v2f b0 = { ldm(0),  ldm(1)  };
        v2f b1 = { ldm(4),  ldm(5)  };
        v2f b2 = { ldm(8),  ldm(9)  };
        v2f b3 = { ldm(12), ldm(13) };
        c0 = __builtin_amdgcn_wmma_f32_16x16x4_f32(false, a, false, b0, (short)0, c0, false, false);
        c1 = __builtin_amdgcn_wmma_f32_16x16x4_f32(false, a, false, b1, (short)0, c1, false, false);
        c2 = __builtin_amdgcn_wmma_f32_16x16x4_f32(false, a, false, b2, (short)0, c2, false, false);
        c3 = __builtin_amdgcn_wmma_f32_16x16x4_f32(false, a, false, b3, (short)0, c3, false, false);
    }

    // D VGPR0 holds the column sum for N = lane&15 in every lane (all M rows
    // identical since A is all ones). Store the strip from lanes 0-15.
    float sum = (c0[0] + c1[0]) + (c2[0] + c3[0]);
    if (valid && lane < 16) {
        float* o = out + (size_t)(tail ? B : 0) * strideH
                       + (size_t)b * strideH + col;
        *o = sum / (float)count;
    }
}

extern "C" void kernel_launch(void* const* d_in, const int* in_sizes, int n_in,
                              void* d_out, int out_size, void* d_ws, size_t ws_size,
                              hipStream_t stream) {
    const float* seq  = (const float*)d_in[0];
    const int*   epos = (const int*)d_in[1];
    float*       out  = (float*)d_out;

    const int B = in_sizes[1] / 4;                                      // 128
    const int H = out_size / (2 * B);                                   // 768
    const int L = (int)((size_t)in_sizes[0] / ((size_t)B * (size_t)H)); // 2048

    dim3 block(256, 1, 1);                        // 8 waves -> 128 columns
    dim3 grid((H + 127) / 128, 2 * B, 1);         // (6, 256) for the ref shapes

    if (H == 768) {
        span_mean_wmma_kernel<768><<<grid, block, 0, stream>>>(seq, epos, out, B, L, H);
    } else {
        span_mean_wmma_kernel<0><<<grid, block, 0, stream>>>(seq, epos, out, B, L, H);
    }
}